// RBPNN_33217277067568
// MI455X (gfx1250) — compile-verified
//
#include <hip/hip_runtime.h>
#include <hip/hip_bf16.h>

// Problem sizes (fixed by the reference)
#define BATCH 4096
#define FEAT  128
#define CTOT  16384
#define OUTT  512
#define CSPLIT 4            // split centre range across grid.y, merge via global atomics
#define MBLK  32            // x rows per workgroup (2 WMMA M-tiles)
#define PAD   132           // padded LDS row stride (floats); 132*4=528B = 33*16B aligned
#define RBF_THREADS 128     // 4 waves -> per-wave double-buffered centre tiles
#define RBF_WAVES   4

typedef __attribute__((ext_vector_type(2))) float v2f;
typedef __attribute__((ext_vector_type(4))) float v4f;
typedef __attribute__((ext_vector_type(8))) float v8f;
typedef __attribute__((ext_vector_type(4))) int   v4i;

#if __has_builtin(__builtin_amdgcn_global_load_async_to_lds_b128)
#define HAVE_ASYNC_LDS 1
typedef __attribute__((address_space(1))) v4i* gas_b128_t;  // global b128 payload ptr
typedef __attribute__((address_space(3))) v4i* las_b128_t;  // LDS b128 payload ptr
#else
#define HAVE_ASYNC_LDS 0
#endif

// ---------------------------------------------------------------------------
// Prep 0: zero the rbf_sum accumulator workspace (poisoned by harness)
// ---------------------------------------------------------------------------
__global__ __launch_bounds__(256) void k_zero(float* __restrict__ p, int n4) {
  int i = blockIdx.x * blockDim.x + threadIdx.x;
  if (i < n4) {
    v4f z = {0.f, 0.f, 0.f, 0.f};
    *(v4f*)(p + 4 * i) = z;
  }
}

// ---------------------------------------------------------------------------
// Prep 1: row norms.  One wave32 per 128-float row; shuffle reduction.
// rows [0,BATCH) -> xnorm ; rows [BATCH, BATCH+CTOT) -> cnorm + cscale
// ---------------------------------------------------------------------------
__global__ __launch_bounds__(256) void k_norms(const float* __restrict__ x,
                                               const float* __restrict__ centres,
                                               const float* __restrict__ log_sigmas,
                                               float* __restrict__ xnorm,
                                               float* __restrict__ cnorm,
                                               float* __restrict__ cscale) {
  const int wave = threadIdx.x >> 5;
  const int lane = threadIdx.x & 31;
  const int row  = blockIdx.x * 8 + wave;
  if (row >= BATCH + CTOT) return;

  const float* src = (row < BATCH) ? (x + (size_t)row * FEAT)
                                   : (centres + (size_t)(row - BATCH) * FEAT);
  v4f v = *(const v4f*)(src + lane * 4);
  float s = v.x * v.x + v.y * v.y + v.z * v.z + v.w * v.w;
  #pragma unroll
  for (int off = 16; off > 0; off >>= 1) s += __shfl_xor(s, off, 32);

  if (lane == 0) {
    if (row < BATCH) {
      xnorm[row] = s;
    } else {
      int c = row - BATCH;
      cnorm[c]  = s;
      cscale[c] = __expf(2.f * log_sigmas[c]);
    }
  }
}

// ---------------------------------------------------------------------------
// Centre-tile staging: 16 rows x 128 floats into a padded LDS slot.
// Async path: 16 x GLOBAL_LOAD_ASYNC_TO_LDS_B128 per wave (ASYNCcnt-tracked),
// one b128 per lane per row.  Sync fallback: b128 load + ds_store.
// ---------------------------------------------------------------------------
__device__ __forceinline__ void stage_tile_async(const float* __restrict__ gsrc,
                                                 float* __restrict__ ldst,
                                                 int lane) {
#if HAVE_ASYNC_LDS
  #pragma unroll
  for (int r = 0; r < 16; ++r) {
    const float* g = gsrc + r * FEAT + lane * 4;
    float*       l = ldst + r * PAD  + lane * 4;
    __builtin_amdgcn_global_load_async_to_lds_b128(
        (gas_b128_t)g, (las_b128_t)l, /*offset=*/0, /*cpol=*/0);
  }
#else
  #pragma unroll
  for (int r = 0; r < 16; ++r) {
    v4f v = *(const v4f*)(gsrc + r * FEAT + lane * 4);
    *(v4f*)(ldst + r * PAD + lane * 4) = v;
  }
#endif
}

__device__ __forceinline__ void wait_async_le16() {
#if HAVE_ASYNC_LDS
  asm volatile("s_wait_asynccnt 16" ::: "memory");
#endif
}
__device__ __forceinline__ void wait_async_le0() {
#if HAVE_ASYNC_LDS
  asm volatile("s_wait_asynccnt 0" ::: "memory");
#endif
}

// ---------------------------------------------------------------------------
// Fused: distance GEMM (f32 WMMA) + exp + segment-sum into LDS accumulator.
// Grid: (BATCH/MBLK, CSPLIT), block 128 (4 waves), double-buffered async
// centre staging overlapped with the WMMA K-loop.
// ---------------------------------------------------------------------------
__global__ __launch_bounds__(RBF_THREADS) void k_rbf(
    const float* __restrict__ x,
    const float* __restrict__ centres,
    const int*   __restrict__ node_id,
    const float* __restrict__ xnorm,
    const float* __restrict__ cnorm,
    const float* __restrict__ cscale,
    float* __restrict__ rbf) {
  __shared__ float xs[MBLK * PAD];                      //  16.5 KB  staged x block
  __shared__ float cs[RBF_WAVES * 2 * 16 * PAD];        //  66.0 KB  double-buffered tiles
  __shared__ float racc[MBLK * OUTT];                   //  64.0 KB  segment-sum accumulator

  const int tid  = threadIdx.x;
  const int wave = tid >> 5;
  const int lane = tid & 31;
  const int hi   = lane >> 4;               // half-wave (K split)
  const int lo   = lane & 15;               // M (for A) / N (for B) index
  const int mbase = blockIdx.x * MBLK;
  const int cbase = blockIdx.y * (CTOT / CSPLIT);

  // zero LDS accumulator
  #pragma unroll 8
  for (int i = 0; i < (MBLK * OUTT) / RBF_THREADS; ++i) racc[tid + i * RBF_THREADS] = 0.f;

  // stage x block: 32 rows x 128 floats, coalesced float4
  #pragma unroll
  for (int j = 0; j < 8; ++j) {
    int i    = tid + j * RBF_THREADS;       // float4 index, 1024 total
    int flat = i * 4;
    int r    = flat >> 7;
    int k    = flat & 127;
    v4f v = *(const v4f*)(x + (size_t)(mbase + r) * FEAT + k);
    *(v4f*)(&xs[r * PAD + k]) = v;
  }
  __syncthreads();

  // per-lane x-norms for the two M-tiles (register resident)
  float xn0[8], xn1[8];
  #pragma unroll
  for (int v = 0; v < 8; ++v) {
    xn0[v] = xnorm[mbase + v + 8 * hi];
    xn1[v] = xnorm[mbase + 16 + v + 8 * hi];
  }

  float* buf0 = &cs[(wave * 2 + 0) * 16 * PAD];         // wave-private double buffer
  float* buf1 = &cs[(wave * 2 + 1) * 16 * PAD];
  const int NTILES = (CTOT / CSPLIT) / 16;              // 256 tiles; 64 per wave, uniform

  // prologue: stage first tile
  stage_tile_async(centres + (size_t)(cbase + wave * 16) * FEAT, buf0, lane);

  int parity = 0;
  for (int nt = wave; nt < NTILES; nt += RBF_WAVES) {
    const int c0  = cbase + nt * 16;
    const int ntn = nt + RBF_WAVES;
    const bool more = (ntn < NTILES);                   // wave-uniform

    // kick off next tile into the other buffer, then wait for current one.
    if (more) {
      stage_tile_async(centres + (size_t)(cbase + ntn * 16) * FEAT,
                       parity ? buf0 : buf1, lane);
      wait_async_le16();      // async done in-order: <=16 left => current tile landed
    } else {
      wait_async_le0();
    }
    float* myc = parity ? buf1 : buf0;

    // per-lane column metadata (n = lo)
    const int   cg = c0 + lo;
    const float cn = cnorm[cg];
    const float sc = cscale[cg];
    const int   nd = node_id[cg];

    v8f acc0 = {}, acc1 = {};
    // K = 128 -> 32 chained V_WMMA_F32_16X16X4_F32 per M-tile, B reused
    #pragma unroll 8
    for (int kk = 0; kk < FEAT / 4; ++kk) {
      const int kb = kk * 4 + 2 * hi;       // A/B lane layout: K = 2*hi + vgpr
      v2f a0 = *(const v2f*)(&xs[lo * PAD + kb]);
      v2f a1 = *(const v2f*)(&xs[(16 + lo) * PAD + kb]);
      v2f bt = *(const v2f*)(&myc[lo * PAD + kb]);
      acc0 = __builtin_amdgcn_wmma_f32_16x16x4_f32(false, a0, false, bt,
                                                   (short)0, acc0, false, false);
      acc1 = __builtin_amdgcn_wmma_f32_16x16x4_f32(false, a1, false, bt,
                                                   (short)0, acc1, false, false);
    }

    // epilogue: sqd -> phi -> segment scatter-add (LDS atomics)
    #pragma unroll
    for (int v = 0; v < 8; ++v) {
      float sq0 = fmaxf(xn0[v] + cn - 2.f * acc0[v], 0.f);
      float sq1 = fmaxf(xn1[v] + cn - 2.f * acc1[v], 0.f);
      float p0 = __expf(-sq0 * sc);
      float p1 = __expf(-sq1 * sc);
      atomicAdd(&racc[(v + 8 * hi) * OUTT + nd], p0);
      atomicAdd(&racc[(16 + v + 8 * hi) * OUTT + nd], p1);
    }
    parity ^= 1;
  }
  __syncthreads();

  // flush partial sums (CSPLIT blocks share each row) via global f32 atomics
  #pragma unroll 8
  for (int i = 0; i < (MBLK * OUTT) / RBF_THREADS; ++i) {
    int idx = tid + i * RBF_THREADS;
    unsafeAtomicAdd(&rbf[(size_t)mbase * OUTT + idx], racc[idx]);
  }
}

// ---------------------------------------------------------------------------
// out = rbf_sum @ W^T + b.  One wave per 16x16 output tile, K=512 f32 WMMA.
// Grid: (BATCH/16)*(OUTT/16)/8 blocks of 256 threads.
// ---------------------------------------------------------------------------
__global__ __launch_bounds__(256) void k_out(const float* __restrict__ rbf,
                                             const float* __restrict__ W,
                                             const float* __restrict__ bias,
                                             float* __restrict__ out) {
  const int wave = threadIdx.x >> 5;
  const int lane = threadIdx.x & 31;
  const int hi = lane >> 4, lo = lane & 15;
  const int tile  = blockIdx.x * 8 + wave;      // 8192 tiles total
  const int mt    = tile >> 5;                  // 32 N-tiles per M row
  const int nt    = tile & 31;
  const int mbase = mt * 16, nbase = nt * 16;

  const float* arow = rbf + (size_t)(mbase + lo) * OUTT;  // A: rbf rows
  const float* brow = W   + (size_t)(nbase + lo) * OUTT;  // B: W rows (= W^T cols)

  v8f acc = {};
  #pragma unroll 8
  for (int kk = 0; kk < OUTT / 4; ++kk) {
    const int kb = kk * 4 + 2 * hi;
    v2f a = *(const v2f*)(arow + kb);
    v2f b = *(const v2f*)(brow + kb);
    acc = __builtin_amdgcn_wmma_f32_16x16x4_f32(false, a, false, b,
                                                (short)0, acc, false, false);
  }

  const float bv = bias[nbase + lo];
  #pragma unroll
  for (int v = 0; v < 8; ++v)
    out[(size_t)(mbase + v + 8 * hi) * OUTT + nbase + lo] = acc[v] + bv;
}

// ---------------------------------------------------------------------------
extern "C" void kernel_launch(void* const* d_in, const int* in_sizes, int n_in,
                              void* d_out, int out_size, void* d_ws, size_t ws_size,
                              hipStream_t stream) {
  const float* x          = (const float*)d_in[0];
  const float* centres    = (const float*)d_in[1];
  const float* log_sigmas = (const float*)d_in[2];
  const int*   node_id    = (const int*)  d_in[3];
  const float* W          = (const float*)d_in[4];
  const float* b          = (const float*)d_in[5];
  float*       out        = (float*)d_out;

  // workspace layout
  float* rbf    = (float*)d_ws;                 // [BATCH, OUTT]
  float* xnorm  = rbf   + (size_t)BATCH * OUTT; // [BATCH]
  float* cnorm  = xnorm + BATCH;                // [CTOT]
  float* cscale = cnorm + CTOT;                 // [CTOT]

  // 1) zero rbf accumulator
  {
    int n4 = (BATCH * OUTT) / 4;
    k_zero<<<(n4 + 255) / 256, 256, 0, stream>>>(rbf, n4);
  }
  // 2) row norms + sigma scales
  {
    int rows = BATCH + CTOT;
    k_norms<<<(rows + 7) / 8, 256, 0, stream>>>(x, centres, log_sigmas,
                                                xnorm, cnorm, cscale);
  }
  // 3) fused distance-GEMM + RBF + segment-sum (async double-buffered staging)
  {
    dim3 grid(BATCH / MBLK, CSPLIT);
    k_rbf<<<grid, RBF_THREADS, 0, stream>>>(x, centres, node_id,
                                            xnorm, cnorm, cscale, rbf);
  }
  // 4) output GEMM + bias
  {
    int tiles = (BATCH / 16) * (OUTT / 16);     // 8192
    k_out<<<tiles / 8, 256, 0, stream>>>(rbf, W, b, out);
  }
}